// Wavenet_76562087018950
// MI455X (gfx1250) — compile-verified
//
#include <hip/hip_runtime.h>
#include <stdint.h>

#define N_LAYERS 30
#define TAPS 5
#define NTHREADS 1024
#define NITER 5
#define EXT (NTHREADS * NITER)        // 5120
#define HALO 576                      // 4 (causal) + 4 * sum(dilations) = 4 + 4*143
#define CHUNK (EXT - HALO)            // 4544
#define LPAD 32                       // max tap reach = 4*8 dilated samples
#define LOG2E 1.4426950408889634f

// ---------- fast math (gfx1250-native where available) ----------
__device__ __forceinline__ float fast_exp2(float x) {
#if __has_builtin(__builtin_amdgcn_exp2f)
  return __builtin_amdgcn_exp2f(x);
#else
  return exp2f(x);
#endif
}
__device__ __forceinline__ float fast_rcp(float x) {
#if __has_builtin(__builtin_amdgcn_rcpf)
  return __builtin_amdgcn_rcpf(x);
#else
  return 1.0f / x;
#endif
}
__device__ __forceinline__ float fast_tanh(float x) {
#if __has_builtin(__builtin_amdgcn_tanhf)
  return __builtin_amdgcn_tanhf(x);            // CDNA5 v_tanh_f32
#elif __has_builtin(__builtin_amdgcn_tanh_f32)
  return __builtin_amdgcn_tanh_f32(x);
#else
  float e = fast_exp2(x * 2.8853900817779268f); // exp(2x)
  return 1.0f - 2.0f * fast_rcp(e + 1.0f);
#endif
}
__device__ __forceinline__ float fast_sigmoid(float x) {
  return fast_rcp(1.0f + fast_exp2(x * -LOG2E));
}
__device__ __forceinline__ void wait_asynccnt0() {
#if __has_builtin(__builtin_amdgcn_s_wait_asynccnt)
  __builtin_amdgcn_s_wait_asynccnt(0);
#else
  asm volatile("s_wait_asynccnt 0" ::: "memory");
#endif
}
// CDNA5 async global->LDS (ASYNCcnt-tracked), 16 bytes per lane.
__device__ __forceinline__ void async_load_b128(uint32_t lds_byte, const float* gsrc) {
  asm volatile("global_load_async_to_lds_b128 %0, %1, off"
               :: "v"(lds_byte), "v"(gsrc) : "memory");
}

// ---------- one gated residual layer, compile-time dilation D ----------
template <int D>
__device__ __forceinline__ void layer_step(
    const float* __restrict__ cur, float* __restrict__ nxt, int k,
    float* h_r, float* skip_r,
    const float* s_fw, const float* s_gw, const float* s_fb,
    const float* s_gb, const float* s_rw, const float* s_rb, int tid)
{
  const float fw0 = s_fw[k*TAPS+0], fw1 = s_fw[k*TAPS+1], fw2 = s_fw[k*TAPS+2],
              fw3 = s_fw[k*TAPS+3], fw4 = s_fw[k*TAPS+4];
  const float gw0 = s_gw[k*TAPS+0], gw1 = s_gw[k*TAPS+1], gw2 = s_gw[k*TAPS+2],
              gw3 = s_gw[k*TAPS+3], gw4 = s_gw[k*TAPS+4];
  const float fbk = s_fb[k], gbk = s_gb[k], rwk = s_rw[k], rbk = s_rb[k];
#pragma unroll
  for (int m = 0; m < NITER; ++m) {
    const int i = tid + m * NTHREADS;
    const float h0 = h_r[m];                 // own value: kept in register
    const float v1 = cur[i - 1 * D];         // immediate-offset ds loads
    const float v2 = cur[i - 2 * D];
    const float v3 = cur[i - 3 * D];
    const float v4 = cur[i - 4 * D];
    const float af = fbk + fw4*h0 + fw3*v1 + fw2*v2 + fw1*v3 + fw0*v4;
    const float ag = gbk + gw4*h0 + gw3*v1 + gw2*v2 + gw1*v3 + gw0*v4;
    const float z  = fast_tanh(af) * fast_sigmoid(ag);
    skip_r[m] += z;
    const float hn = fmaf(z, rwk, rbk) + h0;
    h_r[m] = hn;
    nxt[i] = hn;
  }
  __syncthreads();
}

// ---------- Kernel A: full WaveNet stack per tile (state in LDS + regs) ----------
__global__ __launch_bounds__(NTHREADS) void wavenet_stack_kernel(
    const float* __restrict__ x,
    const float* __restrict__ cw, const float* __restrict__ cb,
    const float* __restrict__ fw, const float* __restrict__ fb,
    const float* __restrict__ gw, const float* __restrict__ gb,
    const float* __restrict__ rw, const float* __restrict__ rb,
    const float* __restrict__ c1w, const float* __restrict__ c1b,
    const float* __restrict__ c2w, const float* __restrict__ c2b,
    float* __restrict__ vout, int T)
{
  __shared__ __align__(16) float bufA_s[LPAD + EXT];
  __shared__ __align__(16) float bufB_s[LPAD + EXT];
  __shared__ float s_fw[N_LAYERS * TAPS], s_gw[N_LAYERS * TAPS];
  __shared__ float s_fb[N_LAYERS], s_gb[N_LAYERS], s_rw[N_LAYERS], s_rb[N_LAYERS];
  __shared__ float s_cw[TAPS + 1], s_fin[4];

  float* A = bufA_s + LPAD;
  float* B = bufB_s + LPAD;

  const int tid = threadIdx.x;
  const int row = blockIdx.y;
  const int c0  = blockIdx.x * CHUNK;   // first output sample of this tile
  const int t0  = c0 - HALO;            // global t of A[0]
  const float* xrow = x + (size_t)row * T;

  if (tid < N_LAYERS * TAPS) { s_fw[tid] = fw[tid]; s_gw[tid] = gw[tid]; }
  if (tid < N_LAYERS) { s_fb[tid] = fb[tid]; s_gb[tid] = gb[tid];
                        s_rw[tid] = rw[tid]; s_rb[tid] = rb[tid]; }
  if (tid < TAPS) s_cw[tid] = cw[tid];
  if (tid == TAPS) s_cw[TAPS] = cb[0];
  if (tid == 0) { s_fin[0] = c1w[0]; s_fin[1] = c1b[0]; s_fin[2] = c2w[0]; s_fin[3] = c2b[0]; }
  if (tid < LPAD) { bufA_s[tid] = 0.0f; bufB_s[tid] = 0.0f; }   // zero pads, never rewritten

  // ---- tile load: async 16B transfers where fully in range, scalar fill at edges
  {
    const int nvec = EXT / 4;
    for (int v = tid; v < nvec; v += NTHREADS) {
      const int i = v * 4;
      const int t = t0 + i;
      if (t >= 0 && t + 3 < T) {
        async_load_b128((uint32_t)(uintptr_t)&A[i], xrow + t);
      } else {
#pragma unroll
        for (int j = 0; j < 4; ++j) {
          const int tj = t + j;
          A[i + j] = (tj >= 0 && tj < T) ? xrow[tj] : 0.0f;
        }
      }
    }
    wait_asynccnt0();
  }
  __syncthreads();

  float h_r[NITER], skip_r[NITER];

  // ---- causal conv (d=1, K=5): y[t] = sum_j w[j] * x[t-(4-j)]
  const float cw0 = s_cw[0], cw1 = s_cw[1], cw2 = s_cw[2], cw3 = s_cw[3],
              cw4 = s_cw[4], cbv = s_cw[TAPS];
#pragma unroll
  for (int m = 0; m < NITER; ++m) {
    const int i = tid + m * NTHREADS;
    const float acc = cbv + cw4 * A[i] + cw3 * A[i-1] + cw2 * A[i-2]
                          + cw1 * A[i-3] + cw0 * A[i-4];
    h_r[m] = acc;
    B[i] = acc;
    skip_r[m] = 0.0f;
  }
  __syncthreads();

  // ---- 30 layers: dilations = 1, then cycle {2,4,8,6}, ending with 2 (k=29)
  layer_step<1>(B, A, 0, h_r, skip_r, s_fw, s_gw, s_fb, s_gb, s_rw, s_rb, tid);
  for (int c = 0; c < 7; ++c) {
    const int k = 1 + 4 * c;
    layer_step<2>(A, B, k + 0, h_r, skip_r, s_fw, s_gw, s_fb, s_gb, s_rw, s_rb, tid);
    layer_step<4>(B, A, k + 1, h_r, skip_r, s_fw, s_gw, s_fb, s_gb, s_rw, s_rb, tid);
    layer_step<8>(A, B, k + 2, h_r, skip_r, s_fw, s_gw, s_fb, s_gb, s_rw, s_rb, tid);
    layer_step<6>(B, A, k + 3, h_r, skip_r, s_fw, s_gw, s_fb, s_gb, s_rw, s_rb, tid);
  }
  layer_step<2>(A, B, 29, h_r, skip_r, s_fw, s_gw, s_fb, s_gb, s_rw, s_rb, tid);

  // ---- head: relu*w+b twice, then mu-law expand; write pre-softmax value
  const float w1 = s_fin[0], b1 = s_fin[1], w2 = s_fin[2], b2 = s_fin[3];
#pragma unroll
  for (int m = 0; m < NITER; ++m) {
    const int i = tid + m * NTHREADS;
    if (i >= HALO) {
      const int t = c0 + (i - HALO);
      if (t < T) {
        float v = skip_r[m];
        v = fmaxf(v, 0.0f) * w1 + b1;
        v = fmaxf(v, 0.0f) * w2 + b2;
        const float a = fabsf(v);
        const float mm = (fast_exp2(a * 8.0f) - 1.0f) * (1.0f / 255.0f); // (256^a-1)/255
        vout[(size_t)row * T + t] = (v < 0.0f) ? -mm : mm;
      }
    }
  }
}

// ---------- Kernel B1: per-slice partial softmax stats (deterministic) ----------
// slice = T / PB must be divisible by 256*VPT (holds for T=262144, PB=128: slice=2048)
#define PB 128
#define VPT 2   // float4 loads per thread
__global__ __launch_bounds__(256) void partial_stats_kernel(
    const float* __restrict__ v, float* __restrict__ partials, int T)
{
  __shared__ float red[256];
  const int b = blockIdx.y, p = blockIdx.x, tid = threadIdx.x;
  const int slice = T / PB;
  const float* base = v + (size_t)b * T + (size_t)p * slice;
  const float4* base4 = (const float4*)base;
  const int nvec = slice / 4;   // 512

  float4 vals[VPT];
#pragma unroll
  for (int j = 0; j < VPT; ++j) {
    const int idx = tid + j * 256;
    vals[j] = (idx < nvec) ? base4[idx] : make_float4(-3.4e38f, -3.4e38f, -3.4e38f, -3.4e38f);
  }

  float m = -3.4e38f;
#pragma unroll
  for (int j = 0; j < VPT; ++j)
    m = fmaxf(m, fmaxf(fmaxf(vals[j].x, vals[j].y), fmaxf(vals[j].z, vals[j].w)));
  red[tid] = m; __syncthreads();
  for (int s = 128; s > 0; s >>= 1) {
    if (tid < s) red[tid] = fmaxf(red[tid], red[tid + s]);
    __syncthreads();
  }
  const float mx = red[0]; __syncthreads();

  float sum = 0.0f;
#pragma unroll
  for (int j = 0; j < VPT; ++j) {
    const int idx = tid + j * 256;
    if (idx < nvec) {
      sum += fast_exp2((vals[j].x - mx) * LOG2E);
      sum += fast_exp2((vals[j].y - mx) * LOG2E);
      sum += fast_exp2((vals[j].z - mx) * LOG2E);
      sum += fast_exp2((vals[j].w - mx) * LOG2E);
    }
  }
  red[tid] = sum; __syncthreads();
  for (int s = 128; s > 0; s >>= 1) {
    if (tid < s) red[tid] += red[tid + s];
    __syncthreads();
  }
  if (tid == 0) {
    partials[(size_t)(b * PB + p) * 2 + 0] = mx;
    partials[(size_t)(b * PB + p) * 2 + 1] = red[0];
  }
}

// ---------- Kernel B2: merge partials per row (online-softmax combine) ----------
__global__ __launch_bounds__(PB) void merge_stats_kernel(
    const float* __restrict__ partials, float* __restrict__ stats)
{
  __shared__ float sm[PB], ss[PB];
  const int b = blockIdx.x, tid = threadIdx.x;
  sm[tid] = partials[(size_t)(b * PB + tid) * 2 + 0];
  ss[tid] = partials[(size_t)(b * PB + tid) * 2 + 1];
  __syncthreads();
  for (int s = PB / 2; s > 0; s >>= 1) {
    if (tid < s) {
      const float m1 = sm[tid], s1 = ss[tid];
      const float m2 = sm[tid + s], s2 = ss[tid + s];
      const float mo = fmaxf(m1, m2);
      sm[tid] = mo;
      ss[tid] = s1 * fast_exp2((m1 - mo) * LOG2E) + s2 * fast_exp2((m2 - mo) * LOG2E);
    }
    __syncthreads();
  }
  if (tid == 0) { stats[2 * b] = sm[0]; stats[2 * b + 1] = ss[0]; }
}

// ---------- Kernel C: in-place softmax normalize (float4, 2-D grid) ----------
__global__ __launch_bounds__(256) void softmax_norm_kernel(
    float* __restrict__ v, const float* __restrict__ stats, int T)
{
  const int b = blockIdx.y;
  const int i4 = blockIdx.x * 256 + threadIdx.x;   // float4 index within row
  const int nvec = T / 4;
  if (i4 >= nvec) return;
  const float mx = stats[2 * b];
  const float rs = fast_rcp(stats[2 * b + 1]);
  float4* row4 = (float4*)(v + (size_t)b * T);
  float4 val = row4[i4];
  val.x = fast_exp2((val.x - mx) * LOG2E) * rs;
  val.y = fast_exp2((val.y - mx) * LOG2E) * rs;
  val.z = fast_exp2((val.z - mx) * LOG2E) * rs;
  val.w = fast_exp2((val.w - mx) * LOG2E) * rs;
  row4[i4] = val;
}

extern "C" void kernel_launch(void* const* d_in, const int* in_sizes, int n_in,
                              void* d_out, int out_size, void* d_ws, size_t ws_size,
                              hipStream_t stream) {
  const float* x   = (const float*)d_in[0];
  const float* cw  = (const float*)d_in[1];
  const float* cb  = (const float*)d_in[2];
  const float* fw  = (const float*)d_in[3];
  const float* fb  = (const float*)d_in[4];
  const float* gw  = (const float*)d_in[5];
  const float* gb  = (const float*)d_in[6];
  const float* rw  = (const float*)d_in[7];
  const float* rb  = (const float*)d_in[8];
  const float* c1w = (const float*)d_in[9];
  const float* c1b = (const float*)d_in[10];
  const float* c2w = (const float*)d_in[11];
  const float* c2b = (const float*)d_in[12];
  float* out      = (float*)d_out;
  float* stats    = (float*)d_ws;              // 16 floats
  float* partials = (float*)d_ws + 16;         // 8 * PB * 2 floats

  const int B = 8;
  const int T = in_sizes[0] / B;
  const int nchunks = (T + CHUNK - 1) / CHUNK;

  hipLaunchKernelGGL(wavenet_stack_kernel, dim3(nchunks, B), dim3(NTHREADS), 0, stream,
                     x, cw, cb, fw, fb, gw, gb, rw, rb, c1w, c1b, c2w, c2b, out, T);
  hipLaunchKernelGGL(partial_stats_kernel, dim3(PB, B), dim3(256), 0, stream,
                     out, partials, T);
  hipLaunchKernelGGL(merge_stats_kernel, dim3(B), dim3(PB), 0, stream,
                     partials, stats);
  hipLaunchKernelGGL(softmax_norm_kernel, dim3((T / 4 + 255) / 256, B), dim3(256), 0, stream,
                     out, stats, T);
}